// level_3_matrix_30502857736459
// MI455X (gfx1250) — compile-verified
//
#include <hip/hip_runtime.h>

// CDNA5 / gfx1250: wave32, fp32 WMMA (V_WMMA_F32_16X16X4_F32).
typedef __attribute__((ext_vector_type(2))) float v2f;
typedef __attribute__((ext_vector_type(8))) float v8f;

#define NF    32          // fields
#define NE    16          // embed dim
#define NPAIR 496         // C(32,2)
#define NSLOT 248         // template edges (half of K32's edges)
#define BPB   8           // batch rows per block (one per wave32)
#define WS    33          // padded LDS row stride of W~ (bank-conflict free)
#define MT    31          // M-tiles of 16 rows (496/16)

// pi: product of 4-cycles on field quadruples; pi(T) = complement(T) below.
constexpr int perm4(int v) { return (v & ~3) | ((v + 1) & 3); }

// lex rank of triple (i,j,j+1) over C(32,3)
constexpr int rowBase(int i, int j) {
  int base = 0;
  for (int a = 0; a < i; ++a) base += (31 - a) * (30 - a) / 2;  // C(31-a,2)
  for (int b = i + 1; b < j; ++b) base += (31 - b);
  return base;
}

struct Tabs {
  int slotA[NSLOT]; int slotB[NSLOT];  // template edge per slot (sorted)
  int rowPack[NPAIR];                  // per W~ row: i | j<<6 | lexbase<<12
  int jmin[MT];                        // per tile: min j over its 16 rows
};

constexpr Tabs makeTabs() {
  Tabs t{};
  bool used[NF][NF] = {};
  int ta[NSLOT] = {}, tb[NSLOT] = {}, key[NSLOT] = {};
  int n = 0;
  // Self-complementary split of K32 under pi: T gets e (and pi^2 e if orbit=4).
  for (int i = 0; i < NF; ++i)
    for (int j = i + 1; j < NF; ++j) {
      if (used[i][j]) continue;
      int e[4][2] = {}; int len = 0; int a = i, b = j;
      for (int s = 0; s < 4; ++s) {
        int lo = a < b ? a : b, hi = a < b ? b : a;
        if (s > 0 && lo == i && hi == j) break;   // orbit closed (size 2)
        e[len][0] = lo; e[len][1] = hi; ++len;
        a = perm4(a); b = perm4(b);
      }
      for (int s = 0; s < len; ++s) used[e[s][0]][e[s][1]] = true;
      ta[n] = e[0][0]; tb[n] = e[0][1]; ++n;
      if (len == 4) { ta[n] = e[2][0]; tb[n] = e[2][1]; ++n; }
    }
  // Sort slots by min(j_half0, j_half1) descending -> better zero-block elision.
  for (int s = 0; s < NSLOT; ++s) {
    int pa = perm4(ta[s]), pb = perm4(tb[s]);
    int j1 = pa > pb ? pa : pb;
    key[s] = tb[s] < j1 ? tb[s] : j1;
  }
  for (int s = 1; s < NSLOT; ++s) {
    int ka = key[s], va = ta[s], vb = tb[s]; int q = s - 1;
    while (q >= 0 && key[q] < ka) {
      key[q+1] = key[q]; ta[q+1] = ta[q]; tb[q+1] = tb[q]; --q;
    }
    key[q+1] = ka; ta[q+1] = va; tb[q+1] = vb;
  }
  // Row tables: row 16m+r = slot edge (half 0); row 16m+8+r = pi(edge) (half 1).
  for (int m = 0; m < MT; ++m) {
    int jm = 63;
    for (int r = 0; r < 8; ++r) {
      int s = m * 8 + r;
      int a0 = ta[s], b0 = tb[s];
      int a1 = perm4(a0), b1 = perm4(b0);
      if (a1 > b1) { int tmp = a1; a1 = b1; b1 = tmp; }
      t.slotA[s] = a0; t.slotB[s] = b0;
      t.rowPack[m*16 + r]     = a0 | (b0 << 6) | (rowBase(a0, b0) << 12);
      t.rowPack[m*16 + 8 + r] = a1 | (b1 << 6) | (rowBase(a1, b1) << 12);
      if (b0 < jm) jm = b0;
      if (b1 < jm) jm = b1;
    }
    t.jmin[m] = jm;
  }
  return t;
}
constexpr Tabs TABS = makeTabs();

// out[b] = sum_{p,e} P_b[p,e] * Q_b[p,e],  Q_b = W~ @ x_b  (496x32 @ 32x16),
// W~ rows ordered by TABS so both lane-halves read P from the same registers.
__global__ __launch_bounds__(256) void level3_wmma_kernel(
    const float* __restrict__ x,   // [B, 32, 16] fp32
    const float* __restrict__ w,   // [4960] fp32
    float* __restrict__ out,       // [B, 1] fp32
    int batch) {
  __shared__ float sWt[NPAIR * WS];      // full W~ table, padded (65.5 KB)
  __shared__ float sX[BPB][NF * NE];     // 8 batch rows of x (16 KB)

  const int tid  = threadIdx.x;
  const int lane = tid & 31;
  const int wave = tid >> 5;
  const int half = lane >> 4;            // K-half per 16x4 f32 A/B layout
  const int l15  = lane & 15;

  // ---- build full W~ in LDS from constexpr row table ----
  for (int idx = tid; idx < NPAIR * NF; idx += 256) {
    int p = idx >> 5, k = idx & 31;
    int packed = TABS.rowPack[p];
    int j = (packed >> 6) & 63;
    int base = packed >> 12;
    sWt[p * WS + k] = (k > j) ? w[base + k - j - 1] : 0.0f;
  }

  // ---- stage this block's 8 batch rows of x ----
  const int b0 = blockIdx.x * BPB;
  for (int idx = tid; idx < BPB * NF * NE; idx += 256) {
    int bb = idx / (NF * NE);
    int r  = idx - bb * (NF * NE);
    int gb = b0 + bb; if (gb >= batch) gb = batch - 1;
    sX[bb][r] = x[(size_t)gb * (NF * NE) + r];
  }
  __syncthreads();

  // ---- each wave: one batch row (wave-uniform branch -> EXEC all-ones) ----
  if (b0 + wave < batch) {
    // Lane-private x column, PERMUTED by the lane's half: xv[f] = x[pi^half(f), e].
    // => P products use identical compile-time register indices on both halves.
    const float* xbase = &sX[wave][l15];
    float xv[NF];
    #pragma unroll
    for (int f = 0; f < NF; ++f) {
      int fld = (f & ~3) | ((f + half) & 3);   // half==0: f, half==1: pi(f)
      xv[f] = xbase[fld * NE];
    }

    // B operands (original field order) recovered from permuted xv:
    // half 0 needs x[4ks], x[4ks+1] = xv[4ks], xv[4ks+1]
    // half 1 needs x[4ks+2], x[4ks+3] = xv[4ks+1], xv[4ks+2]   (one-time selects)
    v2f bsel[8];
    #pragma unroll
    for (int ks = 0; ks < 8; ++ks) {
      bsel[ks].x = half ? xv[ks*4 + 1] : xv[ks*4 + 0];
      bsel[ks].y = half ? xv[ks*4 + 2] : xv[ks*4 + 1];
    }

    // Per-lane A base: A layout M = lane&15, K-half = 2*half.
    const float* aRow = sWt + (l15 * WS + 2 * half);

    float acc = 0.0f;
    #pragma unroll
    for (int m = 0; m < MT; ++m) {
      v8f q = {};                        // Q tile accumulator (f32 C/D)
      #pragma unroll
      for (int ks = 0; ks < 8; ++ks) {
        if (TABS.jmin[m] < ks * 4 + 4) { // compile-time: skip all-zero W~ blocks
          v2f a;
          a.x = aRow[m * 16 * WS + ks * 4 + 0];
          a.y = aRow[m * 16 * WS + ks * 4 + 1];
          q = __builtin_amdgcn_wmma_f32_16x16x4_f32(
                  false, a, false, bsel[ks], (short)0, q, false, false);
        }
      }
      // P (.) Q: slot (m,r) -> same xv indices for both halves (no selects).
      #pragma unroll
      for (int r = 0; r < 8; ++r) {
        const int s = m * 8 + r;
        float pv = xv[TABS.slotA[s]] * xv[TABS.slotB[s]];
        acc = fmaf(pv, q[r], acc);
      }
    }

    // wave32 reduction
    for (int off = 16; off > 0; off >>= 1)
      acc += __shfl_xor(acc, off, 32);
    if (lane == 0) out[b0 + wave] = acc;
  }
}

extern "C" void kernel_launch(void* const* d_in, const int* in_sizes, int n_in,
                              void* d_out, int out_size, void* d_ws, size_t ws_size,
                              hipStream_t stream) {
  const float* x = (const float*)d_in[0];   // [B, 32, 16] fp32
  const float* w = (const float*)d_in[1];   // [4960] fp32
  float* out     = (float*)d_out;           // [B, 1] fp32
  const int batch = in_sizes[0] / (NF * NE);
  const int grid  = (batch + BPB - 1) / BPB;
  hipLaunchKernelGGL(level3_wmma_kernel, dim3(grid), dim3(256), 0, stream,
                     x, w, out, batch);
}